// MambaSimulator_12713103196456
// MI455X (gfx1250) — compile-verified
//
#include <hip/hip_runtime.h>
#include <hip/hip_bf16.h>
#include <math.h>

typedef __attribute__((ext_vector_type(2))) float v2f;
typedef __attribute__((ext_vector_type(8))) float v8f;

#define D_MODEL 1024
#define D_STATE 64
#define SEQ     512
#define BATCH   4
#define IN_SIZE 128
#define ROWS    (BATCH * SEQ)   // 2048
#define NSEL    129

// ---------------------------------------------------------------------------
// Kernel 1: U = X @ W_emb + b_emb      (2048x128) x (128x1024) -> (2048x1024)
// One wave per 16x16 output tile; fp32 WMMA, K stepped by 4.
// ---------------------------------------------------------------------------
__global__ __launch_bounds__(256) void k_embed(const float* __restrict__ X,
                                               const float* __restrict__ W,
                                               const float* __restrict__ bias,
                                               float* __restrict__ U)
{
    const int lane  = threadIdx.x & 31;
    const int wave  = threadIdx.x >> 5;
    const int t     = blockIdx.x * 8 + wave;   // 8192 tiles = 128 Mtiles x 64 Ntiles
    const int tileN = t & 63;
    const int tileM = t >> 6;
    const int m0 = tileM * 16, n0 = tileN * 16;
    const int l15   = lane & 15;
    const int khalf = (lane >> 4) << 1;        // 0 or 2
    const int mhigh = (lane >> 4) << 3;        // 0 or 8 (C/D row offset)

    v8f acc;
    const float bv = bias[n0 + l15];
#pragma unroll
    for (int r = 0; r < 8; ++r) acc[r] = bv;

    const float* arow = X + (size_t)(m0 + l15) * IN_SIZE;
    const float* bcol = W + n0 + l15;

#pragma unroll 4
    for (int k = 0; k < IN_SIZE; k += 4) {
        v2f a, b;
        a.x = arow[k + khalf];
        a.y = arow[k + khalf + 1];
        b.x = bcol[(size_t)(k + khalf) * D_MODEL];
        b.y = bcol[(size_t)(k + khalf + 1) * D_MODEL];
        acc = __builtin_amdgcn_wmma_f32_16x16x4_f32(false, a, false, b,
                                                    (short)0, acc, false, false);
    }

#pragma unroll
    for (int r = 0; r < 8; ++r)
        U[(size_t)(m0 + r + mhigh) * D_MODEL + n0 + l15] = acc[r];
}

// ---------------------------------------------------------------------------
// Kernel 2: sel = U @ W_xp + b_xp (129 cols), fused softplus/exp transforms.
// Block per row; U row staged in LDS. Emits A_bar, B_bar, Cs in (row, d) SoA.
// ---------------------------------------------------------------------------
__global__ __launch_bounds__(160) void k_sel(const float* __restrict__ U,
                                             const float* __restrict__ W_xp,
                                             const float* __restrict__ b_xp,
                                             const float* __restrict__ A,
                                             float* __restrict__ Abar,
                                             float* __restrict__ Bbar,
                                             float* __restrict__ Cs)
{
    __shared__ float su[D_MODEL];
    __shared__ float ssel[NSEL];
    const int row = blockIdx.x;
    const int tid = threadIdx.x;

    for (int i = tid; i < D_MODEL; i += 160)
        su[i] = U[(size_t)row * D_MODEL + i];
    __syncthreads();

    if (tid < NSEL) {
        float acc = b_xp[tid];
#pragma unroll 8
        for (int k = 0; k < D_MODEL; ++k)
            acc = fmaf(su[k], W_xp[(size_t)k * NSEL + tid], acc);
        ssel[tid] = acc;
    }
    __syncthreads();

    const float s0    = ssel[0];
    const float sp    = (s0 > 20.f) ? s0 : log1pf(__expf(s0));
    const float delta = 0.01f * sp;

    if (tid < D_STATE) {
        const float ad = A[tid];
        Abar[(size_t)row * D_STATE + tid] = __expf(delta * ad);
        Bbar[(size_t)row * D_STATE + tid] = delta * ssel[1 + tid];
        Cs  [(size_t)row * D_STATE + tid] = ssel[1 + D_STATE + tid];
    }
}

// ---------------------------------------------------------------------------
// Kernel 3: sequential selective scan. Lane owns one (b, m) row; h[64] lives
// entirely in VGPRs. A_bar/B_bar/Cs pointers are wave-uniform (b derived from
// blockIdx only) -> scalar loads; u/y are coalesced b32 per step.
// ---------------------------------------------------------------------------
__global__ __launch_bounds__(256) void k_scan(const float* __restrict__ U,
                                              const float* __restrict__ Abar,
                                              const float* __restrict__ Bbar,
                                              const float* __restrict__ Cs,
                                              const float* __restrict__ Dp,
                                              float* __restrict__ Y)
{
    const int rowbase = blockIdx.x * 256;        // 16 blocks cover 4096 rows
    const int b = rowbase >> 10;                 // uniform per block
    const int m = (rowbase & 1023) + threadIdx.x;

    float h[D_STATE];
#pragma unroll
    for (int d = 0; d < D_STATE; ++d) h[d] = 0.f;

    const float dp = Dp[m];
    const float* Ab = Abar + (size_t)b * SEQ * D_STATE;
    const float* Bb = Bbar + (size_t)b * SEQ * D_STATE;
    const float* Cb = Cs   + (size_t)b * SEQ * D_STATE;
    const float* Ub = U + (size_t)b * SEQ * D_MODEL + m;
    float*       Yb = Y + (size_t)b * SEQ * D_MODEL + m;

    for (int s = 0; s < SEQ; ++s) {
        const float ut = Ub[(size_t)s * D_MODEL];
        const float* a  = Ab + s * D_STATE;
        const float* bb = Bb + s * D_STATE;
        const float* c  = Cb + s * D_STATE;
        float acc = 0.f;
#pragma unroll
        for (int d = 0; d < D_STATE; ++d) {
            const float hd = fmaf(h[d], a[d], bb[d] * ut);
            h[d] = hd;
            acc  = fmaf(hd, c[d], acc);
        }
        Yb[(size_t)s * D_MODEL] = fmaf(ut, dp, acc);
    }
}

// ---------------------------------------------------------------------------
// Kernel 4: LayerNorm over d_model, in-place on Y. Block per row, wave32
// shuffle reduction + cross-wave LDS reduction.
// ---------------------------------------------------------------------------
__global__ __launch_bounds__(256) void k_ln(float* __restrict__ Y,
                                            const float* __restrict__ gamma,
                                            const float* __restrict__ beta)
{
    __shared__ float red0[8];
    __shared__ float red1[8];
    const int row = blockIdx.x;
    float* y = Y + (size_t)row * D_MODEL;

    float v[4];
    float sum = 0.f, sq = 0.f;
#pragma unroll
    for (int i = 0; i < 4; ++i) {
        v[i] = y[threadIdx.x + i * 256];
        sum += v[i];
        sq   = fmaf(v[i], v[i], sq);
    }
#pragma unroll
    for (int off = 16; off > 0; off >>= 1) {
        sum += __shfl_xor(sum, off, 32);
        sq  += __shfl_xor(sq,  off, 32);
    }
    const int wv = threadIdx.x >> 5, ln = threadIdx.x & 31;
    if (ln == 0) { red0[wv] = sum; red1[wv] = sq; }
    __syncthreads();
    if (threadIdx.x == 0) {
        float ts = 0.f, tq = 0.f;
#pragma unroll
        for (int w = 0; w < 8; ++w) { ts += red0[w]; tq += red1[w]; }
        red0[0] = ts; red1[0] = tq;
    }
    __syncthreads();
    const float mu   = red0[0] * (1.f / D_MODEL);
    const float var  = red1[0] * (1.f / D_MODEL) - mu * mu;
    const float rstd = rsqrtf(var + 1e-5f);
#pragma unroll
    for (int i = 0; i < 4; ++i) {
        const int col = threadIdx.x + i * 256;
        y[col] = (v[i] - mu) * rstd * gamma[col] + beta[col];
    }
}

// ---------------------------------------------------------------------------
// Kernel 5: P = Yn @ W_out + b_out    (2048x1024) x (1024x128) -> (2048x128)
// ---------------------------------------------------------------------------
__global__ __launch_bounds__(256) void k_out(const float* __restrict__ Yn,
                                             const float* __restrict__ W,
                                             const float* __restrict__ bias,
                                             float* __restrict__ P)
{
    const int lane  = threadIdx.x & 31;
    const int wave  = threadIdx.x >> 5;
    const int t     = blockIdx.x * 8 + wave;   // 1024 tiles = 128 Mtiles x 8 Ntiles
    const int tileN = t & 7;
    const int tileM = t >> 3;
    const int m0 = tileM * 16, n0 = tileN * 16;
    const int l15   = lane & 15;
    const int khalf = (lane >> 4) << 1;
    const int mhigh = (lane >> 4) << 3;

    v8f acc;
    const float bv = bias[n0 + l15];
#pragma unroll
    for (int r = 0; r < 8; ++r) acc[r] = bv;

    const float* arow = Yn + (size_t)(m0 + l15) * D_MODEL;
    const float* bcol = W + n0 + l15;

#pragma unroll 4
    for (int k = 0; k < D_MODEL; k += 4) {
        v2f a, b;
        a.x = arow[k + khalf];
        a.y = arow[k + khalf + 1];
        b.x = bcol[(size_t)(k + khalf) * IN_SIZE];
        b.y = bcol[(size_t)(k + khalf + 1) * IN_SIZE];
        acc = __builtin_amdgcn_wmma_f32_16x16x4_f32(false, a, false, b,
                                                    (short)0, acc, false, false);
    }

#pragma unroll
    for (int r = 0; r < 8; ++r)
        P[(size_t)(m0 + r + mhigh) * IN_SIZE + n0 + l15] = acc[r];
}

// ---------------------------------------------------------------------------
extern "C" void kernel_launch(void* const* d_in, const int* in_sizes, int n_in,
                              void* d_out, int out_size, void* d_ws, size_t ws_size,
                              hipStream_t stream)
{
    const float* x     = (const float*)d_in[0];
    const float* W_emb = (const float*)d_in[1];
    const float* b_emb = (const float*)d_in[2];
    const float* W_xp  = (const float*)d_in[3];
    const float* b_xp  = (const float*)d_in[4];
    const float* A     = (const float*)d_in[5];
    const float* Dp    = (const float*)d_in[6];
    const float* gamma = (const float*)d_in[7];
    const float* beta  = (const float*)d_in[8];
    const float* W_out = (const float*)d_in[9];
    const float* b_out = (const float*)d_in[10];
    float* out = (float*)d_out;

    // Workspace layout (floats): U (2048x1024), Y (2048x1024),
    // Abar/Bbar/Cs (2048x64 each)  => ~17.5 MB
    float* U    = (float*)d_ws;
    float* Y    = U + (size_t)ROWS * D_MODEL;
    float* Abar = Y + (size_t)ROWS * D_MODEL;
    float* Bbar = Abar + (size_t)ROWS * D_STATE;
    float* Cs   = Bbar + (size_t)ROWS * D_STATE;

    k_embed<<<1024, 256, 0, stream>>>(x, W_emb, b_emb, U);
    k_sel  <<<ROWS, 160, 0, stream>>>(U, W_xp, b_xp, A, Abar, Bbar, Cs);
    k_scan <<<16,   256, 0, stream>>>(U, Abar, Bbar, Cs, Dp, Y);
    k_ln   <<<ROWS, 256, 0, stream>>>(Y, gamma, beta);
    k_out  <<<128,  256, 0, stream>>>(Y, W_out, b_out, out);
}